// GNN_3693671875301
// MI455X (gfx1250) — compile-verified
//
#include <hip/hip_runtime.h>
#include <hip/hip_bf16.h>

#define NH      128
#define NGRAPH  128
#define BN_EPS  1e-5f

typedef __attribute__((ext_vector_type(2))) float v2f;
typedef __attribute__((ext_vector_type(8))) float v8f;

__device__ __forceinline__ float frelu(float v) { return v > 0.f ? v : 0.f; }

// ---------------- node embedding gather: X[n,:] = emb[idx[n],:] ----------------
__global__ __launch_bounds__(256) void k_embed(const int* __restrict__ idx,
                                               const float* __restrict__ emb,
                                               float* __restrict__ X, int n) {
    int node = (int)((blockIdx.x * 256u + threadIdx.x) >> 5);
    int lane = threadIdx.x & 31;
    if (node >= n) return;
    int e = idx[node];
    const float4* s = (const float4*)(emb + (size_t)e * NH);
    float4*       d = (float4*)(X + (size_t)node * NH);
    d[lane] = s[lane];   // 32 lanes x float4 = 128 floats
}

// ---------------- GINE message + scatter-add: H[dst] += relu(X[src] + ea) ------
__global__ __launch_bounds__(256) void k_edge(const float* __restrict__ X,
                                              const int* __restrict__ src,
                                              const int* __restrict__ dst,
                                              const int* __restrict__ eattr,
                                              const float* __restrict__ eemb,
                                              float* __restrict__ H, int ne) {
    int e    = (int)((blockIdx.x * 256u + threadIdx.x) >> 5);   // one wave per edge
    int lane = threadIdx.x & 31;
    if (e >= ne) return;
    int s = src[e], d = dst[e], a = eattr[e];
    float4 xv = ((const float4*)(X + (size_t)s * NH))[lane];
    float4 ev = ((const float4*)(eemb + (size_t)a * NH))[lane];
    float* o = H + (size_t)d * NH + lane * 4;
    unsafeAtomicAdd(o + 0, frelu(xv.x + ev.x));
    unsafeAtomicAdd(o + 1, frelu(xv.y + ev.y));
    unsafeAtomicAdd(o + 2, frelu(xv.z + ev.z));
    unsafeAtomicAdd(o + 3, frelu(xv.w + ev.w));
}

// ---------------- fused GINE MLP: H <- relu(H @ W1) @ W2, + column stats -------
// One block = 8 waves, each wave owns a 16-row strip (128 rows/block).
// WMMA F32 16x16x4: A lane layout m=lane&15, K pair at 2*(lane>>4); B mirrors.
__global__ __launch_bounds__(256) void k_gine_mlp(float* __restrict__ H,
                                                  const float* __restrict__ W1,
                                                  const float* __restrict__ W2,
                                                  float* __restrict__ stats,
                                                  int nrows) {
    extern __shared__ float lds[];
    float* w1s  = lds;                       // 16384 floats
    float* w2s  = lds + 16384;               // 16384 floats
    float* tile = lds + 32768;               // 8 waves * 16*132 floats
    float* sst  = lds + 32768 + 8 * 2112;    // 256 floats (sum | sumsq)

    const int tid  = threadIdx.x;
    const int wave = tid >> 5;
    const int lane = tid & 31;

    // cooperative weight staging (float4)
    for (int i = tid; i < 4096; i += 256) {
        ((float4*)w1s)[i] = ((const float4*)W1)[i];
        ((float4*)w2s)[i] = ((const float4*)W2)[i];
    }
    sst[tid] = 0.f;
    __syncthreads();

    const int m    = lane & 15;
    const int half = lane >> 4;
    const int kh   = half << 1;              // 0 or 2
    const int rowBase = blockIdx.x * 128 + wave * 16;
    const int row  = rowBase + m;
    const bool rv  = row < nrows;
    const float* hrow = H + (size_t)row * NH;
    float* tw = tile + wave * 2112;          // 16 x 132 (padded)

    // A fragments for GEMM1 (kept in registers via full unroll)
    v2f a1[32];
#pragma unroll
    for (int kt = 0; kt < 32; ++kt) {
        v2f a = {0.f, 0.f};
        if (rv) { a.x = hrow[kt * 4 + kh]; a.y = hrow[kt * 4 + kh + 1]; }
        a1[kt] = a;
    }

    // GEMM1: T = relu(H @ W1) -> LDS (transposed access pattern for GEMM2)
    for (int nt = 0; nt < 8; ++nt) {
        v8f acc = {0.f, 0.f, 0.f, 0.f, 0.f, 0.f, 0.f, 0.f};
#pragma unroll
        for (int kt = 0; kt < 32; ++kt) {
            v2f b;
            b.x = w1s[(kt * 4 + kh) * NH + nt * 16 + m];
            b.y = w1s[(kt * 4 + kh + 1) * NH + nt * 16 + m];
            acc = __builtin_amdgcn_wmma_f32_16x16x4_f32(
                false, a1[kt], false, b, (short)0, acc, false, false);
        }
#pragma unroll
        for (int i = 0; i < 8; ++i)
            tw[(i + 8 * half) * 132 + nt * 16 + m] = frelu(acc[i]);
    }
    __syncthreads();

    // A fragments for GEMM2 from LDS tile
    v2f a2[32];
#pragma unroll
    for (int kt = 0; kt < 32; ++kt) {
        v2f a;
        a.x = tw[m * 132 + kt * 4 + kh];
        a.y = tw[m * 132 + kt * 4 + kh + 1];
        a2[kt] = a;
    }

    // GEMM2: Z = T @ W2 -> global (in-place over this block's rows) + stats
    for (int nt = 0; nt < 8; ++nt) {
        v8f acc = {0.f, 0.f, 0.f, 0.f, 0.f, 0.f, 0.f, 0.f};
#pragma unroll
        for (int kt = 0; kt < 32; ++kt) {
            v2f b;
            b.x = w2s[(kt * 4 + kh) * NH + nt * 16 + m];
            b.y = w2s[(kt * 4 + kh + 1) * NH + nt * 16 + m];
            acc = __builtin_amdgcn_wmma_f32_16x16x4_f32(
                false, a2[kt], false, b, (short)0, acc, false, false);
        }
        float s = 0.f, q = 0.f;
#pragma unroll
        for (int i = 0; i < 8; ++i) {
            int r = rowBase + i + 8 * half;
            float v = acc[i];
            if (r < nrows) {
                H[(size_t)r * NH + nt * 16 + m] = v;
                s += v; q += v * v;
            }
        }
        s += __shfl_xor(s, 16);
        q += __shfl_xor(q, 16);
        if (half == 0) {
            atomicAdd(&sst[nt * 16 + m], s);          // ds_add_f32
            atomicAdd(&sst[NH + nt * 16 + m], q);
        }
    }
    __syncthreads();
    unsafeAtomicAdd(&stats[tid], sst[tid]);           // 256 global atomics / block
}

// ---------------- BN (+relu) + residual: X += relu(BN(Z)) ----------------------
__global__ __launch_bounds__(256) void k_bn_res(const float* __restrict__ Z,
                                                float* __restrict__ X,
                                                const float* __restrict__ stats,
                                                const float* __restrict__ gamma,
                                                const float* __restrict__ beta,
                                                float invN, int total) {
    int i = (int)(blockIdx.x * 256u + threadIdx.x);
    if (i >= total) return;
    int c = i & (NH - 1);
    float mu  = stats[c] * invN;
    float var = stats[NH + c] * invN - mu * mu;
    float v = (Z[i] - mu) * rsqrtf(var + BN_EPS) * gamma[c] + beta[c];
    X[i] += fmaxf(v, 0.f);
}

// ---------------- graph pooling: G[batch[n]] += X[n] ---------------------------
__global__ __launch_bounds__(256) void k_pool(const float* __restrict__ X,
                                              const int* __restrict__ batch,
                                              float* __restrict__ G, int n) {
    int node = (int)((blockIdx.x * 256u + threadIdx.x) >> 5);
    int lane = threadIdx.x & 31;
    if (node >= n) return;
    int b = batch[node];
    float4 xv = ((const float4*)(X + (size_t)node * NH))[lane];
    float* g = G + (size_t)b * NH + lane * 4;
    unsafeAtomicAdd(g + 0, xv.x);
    unsafeAtomicAdd(g + 1, xv.y);
    unsafeAtomicAdd(g + 2, xv.z);
    unsafeAtomicAdd(g + 3, xv.w);
}

__global__ void k_zero(float* p, int n) {
    int i = (int)(blockIdx.x * blockDim.x + threadIdx.x);
    if (i < n) p[i] = 0.f;
}

// ---------------- output head: Linear -> BN -> ReLU -> Linear (single block) ---
__global__ __launch_bounds__(256) void k_out(const float* __restrict__ G,
                                             const float* __restrict__ W1,
                                             const float* __restrict__ b1,
                                             const float* __restrict__ gam,
                                             const float* __restrict__ bet,
                                             const float* __restrict__ W2,
                                             const float* __restrict__ b2,
                                             float* __restrict__ out) {
    extern __shared__ float lds[];
    float* o1 = lds;             // 16384 floats
    float* mu = lds + 16384;     // 128
    float* is = lds + 16512;     // 128
    int tid = threadIdx.x;
    for (int idx = tid; idx < NGRAPH * NH; idx += 256) {
        int r = idx >> 7, c = idx & (NH - 1);
        float acc = b1[c];
        for (int k = 0; k < NH; ++k) acc += G[r * NH + k] * W1[k * NH + c];
        o1[idx] = acc;
    }
    __syncthreads();
    if (tid < NH) {
        float s = 0.f, q = 0.f;
        for (int r = 0; r < NGRAPH; ++r) { float v = o1[r * NH + tid]; s += v; q += v * v; }
        float m = s * (1.f / NGRAPH);
        mu[tid] = m;
        is[tid] = rsqrtf(q * (1.f / NGRAPH) - m * m + BN_EPS);
    }
    __syncthreads();
    if (tid < NGRAPH) {
        float acc = b2[0];
        for (int c = 0; c < NH; ++c) {
            float v = (o1[tid * NH + c] - mu[c]) * is[c] * gam[c] + bet[c];
            acc += fmaxf(v, 0.f) * W2[c];
        }
        out[tid] = acc;
    }
}

extern "C" void kernel_launch(void* const* d_in, const int* in_sizes, int n_in,
                              void* d_out, int out_size, void* d_ws, size_t ws_size,
                              hipStream_t stream) {
    const int*   x_idx      = (const int*)d_in[0];
    const int*   edge_index = (const int*)d_in[1];
    const int*   edge_attr  = (const int*)d_in[2];
    const int*   batch      = (const int*)d_in[3];
    const float* node_emb   = (const float*)d_in[4];
    const float* edge_emb   = (const float*)d_in[5];   // [4,20,128]
    const float* conv_w1    = (const float*)d_in[6];   // [4,128,128]
    const float* conv_w2    = (const float*)d_in[7];
    const float* bn_g       = (const float*)d_in[8];   // [4,128]
    const float* bn_b       = (const float*)d_in[9];
    const float* out_w1     = (const float*)d_in[10];
    const float* out_b1     = (const float*)d_in[11];
    const float* obn_g      = (const float*)d_in[12];
    const float* obn_b      = (const float*)d_in[13];
    const float* out_w2     = (const float*)d_in[14];
    const float* out_b2     = (const float*)d_in[15];

    const int N = in_sizes[0];
    const int E = in_sizes[2];
    const int* esrc = edge_index;
    const int* edst = edge_index + E;

    // workspace layout
    float* X     = (float*)d_ws;                 // [N,128]
    float* H     = X + (size_t)N * NH;           // [N,128]
    float* stats = H + (size_t)N * NH;           // [256]
    float* G     = stats + 2 * NH;               // [128,128]

    const size_t nodeBytes = (size_t)N * NH * sizeof(float);
    const int embBlocks  = (N * 32 + 255) / 256;
    const int edgeBlocks = (int)(((long long)E * 32 + 255) / 256);
    const int gemmBlocks = (N + 127) / 128;
    const int bnBlocks   = (N * NH + 255) / 256;
    const size_t ldsGemm = (size_t)(16384 + 16384 + 8 * 2112 + 256) * sizeof(float);
    const size_t ldsOut  = (size_t)(16384 + 256) * sizeof(float);

    k_embed<<<embBlocks, 256, 0, stream>>>(x_idx, node_emb, X, N);

    for (int l = 0; l < 4; ++l) {
        hipMemcpyAsync(H, X, nodeBytes, hipMemcpyDeviceToDevice, stream);  // H = x (self term)
        k_edge<<<edgeBlocks, 256, 0, stream>>>(X, esrc, edst, edge_attr,
                                               edge_emb + (size_t)l * 20 * NH, H, E);
        k_zero<<<1, 256, 0, stream>>>(stats, 2 * NH);
        k_gine_mlp<<<gemmBlocks, 256, ldsGemm, stream>>>(
            H, conv_w1 + (size_t)l * NH * NH, conv_w2 + (size_t)l * NH * NH, stats, N);
        k_bn_res<<<bnBlocks, 256, 0, stream>>>(H, X, stats,
                                               bn_g + l * NH, bn_b + l * NH,
                                               1.f / (float)N, N * NH);
    }

    k_zero<<<(NGRAPH * NH + 255) / 256, 256, 0, stream>>>(G, NGRAPH * NH);
    k_pool<<<(N * 32 + 255) / 256, 256, 0, stream>>>(X, batch, G, N);
    k_out<<<1, 256, ldsOut, stream>>>(G, out_w1, out_b1, obn_g, obn_b,
                                      out_w2, out_b2, (float*)d_out);
}